// SSIM_Loss_9887014716191
// MI455X (gfx1250) — compile-verified
//
#include <hip/hip_runtime.h>
#include <math.h>

typedef float v2f __attribute__((ext_vector_type(2)));
typedef float v8f __attribute__((ext_vector_type(8)));

// Problem constants (reference: B=4, C=3, H=1088, W=1920)
constexpr int IMG_H   = 1088;
constexpr int IMG_W   = 1920;
constexpr int NPLANES = 12;          // B*C
constexpr int TILE_H  = 32;          // per-block output rows (2 wave tiles)
constexpr int TILE_W  = 64;          // per-block output cols (4 wave tiles)
constexpr int EXT_H   = 48;          // 32 + halo + WMMA row padding (needs rows 0..47)
constexpr int EXT_W   = 76;          // 64 + 10 halo + K-padding cols (needs cols 0..75)
constexpr int SH_PITCH = 20;         // per-wave scratch pitch (bank-conflict-free)
constexpr float SSIM_C1 = 0.01f * 0.01f;
constexpr float SSIM_C2 = 0.03f * 0.03f;

template <int F>
__device__ __forceinline__ float fv(float x, float y) {
  if constexpr (F == 0) return x;
  else if constexpr (F == 1) return y;
  else if constexpr (F == 2) return x * x;
  else if constexpr (F == 3) return y * y;
  else return x * y;
}

// Separable 11-tap conv of one field over this wave's 16x16 tile, via
// V_WMMA_F32_16X16X4_F32.  Horizontal: Hout = In(16x28) x Wband(28x16),
// two row-tiles covering extended rows 0..31.  Vertical: Out = Gband(16x28)
// x Hout(28x16).  Both banded weight operands share one per-lane register
// image wq0/wq1 (A and B layouts are mutual transposes).
template <int F>
__device__ __forceinline__ v8f conv_field(const float* __restrict__ s1,
                                          const float* __restrict__ s2,
                                          float* __restrict__ myH,
                                          const float* wq0, const float* wq1,
                                          int wy, int wx, int half, int l) {
#pragma unroll
  for (int t = 0; t < 2; ++t) {
    v8f acc = {};
    const float* r1 = s1 + (wy * 16 + t * 16 + l) * EXT_W + wx * 16;
    const float* r2 = s2 + (wy * 16 + t * 16 + l) * EXT_W + wx * 16;
#pragma unroll
    for (int q = 0; q < 7; ++q) {
      int c0 = 4 * q + 2 * half;                 // A: row = lane&15, K = 4q+2*half+slot
      v2f a, b;
      a.x = fv<F>(r1[c0], r2[c0]);
      a.y = fv<F>(r1[c0 + 1], r2[c0 + 1]);
      b.x = wq0[q];                              // B: banded gaussian chunk
      b.y = wq1[q];
      acc = __builtin_amdgcn_wmma_f32_16x16x4_f32(false, a, false, b,
                                                  (short)0, acc, false, false);
    }
    // D layout: VGPR i holds row i (lanes 0-15) / row i+8 (lanes 16-31)
#pragma unroll
    for (int i = 0; i < 8; ++i)
      myH[(t * 16 + i + 8 * half) * SH_PITCH + l] = acc[i];
  }

  v8f acc = {};
#pragma unroll
  for (int q = 0; q < 7; ++q) {
    int e0 = 4 * q + 2 * half;                   // B: row = 4q+2*half+slot, col = lane&15
    v2f a, b;
    a.x = wq0[q];                                // A: banded gaussian chunk
    a.y = wq1[q];
    b.x = myH[e0 * SH_PITCH + l];
    b.y = myH[(e0 + 1) * SH_PITCH + l];
    acc = __builtin_amdgcn_wmma_f32_16x16x4_f32(false, a, false, b,
                                                (short)0, acc, false, false);
  }
  return acc;
}

__global__ __launch_bounds__(256) void ssim_tile_kernel(
    const float* __restrict__ img1, const float* __restrict__ img2,
    const float* __restrict__ win, float* __restrict__ parts) {
  __shared__ float s1[EXT_H * EXT_W];
  __shared__ float s2[EXT_H * EXT_W];
  __shared__ float sH[8 * 32 * SH_PITCH];        // per-wave Hout scratch
  __shared__ float sG[16];                       // 1D gaussian taps (+zero slots)
  __shared__ float red[256];

  const int tid = threadIdx.x;
  const int bx = blockIdx.x, by = blockIdx.y, p = blockIdx.z;
  const size_t base = (size_t)p * ((size_t)IMG_H * IMG_W);
  const int row0 = by * TILE_H, col0 = bx * TILE_W;

  // Recover 1D kernel from the 2D window input: w2d[i][j] = g[i]*g[j].
  if (tid < 16) {
    float g5 = sqrtf(win[5 * 11 + 5]);
    sG[tid] = (tid < 11) ? (win[5 * 11 + tid] / g5) : 0.0f;
  }

  // Cooperative fill of the zero-padded halo tiles for img1/img2.
  for (int idx = tid; idx < EXT_H * EXT_W; idx += 256) {
    int r = idx / EXT_W, c = idx - r * EXT_W;
    int gr = row0 - 5 + r, gc = col0 - 5 + c;
    float x = 0.0f, y = 0.0f;
    if (gr >= 0 && gr < IMG_H && gc >= 0 && gc < IMG_W) {
      size_t o = base + (size_t)gr * IMG_W + gc;
      x = img1[o];
      y = img2[o];
    }
    s1[idx] = x;
    s2[idx] = y;
  }
  __syncthreads();

  const int lane = tid & 31;
  const int wv = tid >> 5;
  const int wy = wv >> 2, wx = wv & 3;
  const int half = lane >> 4, l = lane & 15;
  float* myH = sH + wv * 32 * SH_PITCH;

  // Per-lane banded-gaussian operand image, shared by A(vertical)/B(horizontal).
  float wq0[7], wq1[7];
#pragma unroll
  for (int q = 0; q < 7; ++q) {
    int k0 = 4 * q + 2 * half - l;
    int k1 = k0 + 1;
    wq0[q] = sG[((unsigned)k0 <= 10u) ? k0 : 11];
    wq1[q] = sG[((unsigned)k1 <= 10u) ? k1 : 11];
  }

  v8f mu1 = conv_field<0>(s1, s2, myH, wq0, wq1, wy, wx, half, l);
  v8f mu2 = conv_field<1>(s1, s2, myH, wq0, wq1, wy, wx, half, l);
  v8f m11 = conv_field<2>(s1, s2, myH, wq0, wq1, wy, wx, half, l);
  v8f m22 = conv_field<3>(s1, s2, myH, wq0, wq1, wy, wx, half, l);
  v8f m12 = conv_field<4>(s1, s2, myH, wq0, wq1, wy, wx, half, l);

  // Pointwise SSIM map + per-thread partial sum.
  float part = 0.0f;
#pragma unroll
  for (int i = 0; i < 8; ++i) {
    float a1 = mu1[i], a2 = mu2[i];
    float a1s = a1 * a1, a2s = a2 * a2, a12 = a1 * a2;
    float sg1 = m11[i] - a1s;
    float sg2 = m22[i] - a2s;
    float sg12 = m12[i] - a12;
    float num = (2.0f * a12 + SSIM_C1) * (2.0f * sg12 + SSIM_C2);
    float den = (a1s + a2s + SSIM_C1) * (sg1 + sg2 + SSIM_C2);
    part += num / den;
  }

  red[tid] = part;
  __syncthreads();
#pragma unroll
  for (int s = 128; s > 0; s >>= 1) {
    if (tid < s) red[tid] += red[tid + s];
    __syncthreads();
  }
  if (tid == 0) {
    int bid = (p * gridDim.y + by) * gridDim.x + bx;
    parts[bid] = red[0];
  }
}

__global__ __launch_bounds__(256) void ssim_reduce_kernel(
    const float* __restrict__ parts, int n, float* __restrict__ out) {
  __shared__ float red[256];
  const int tid = threadIdx.x;
  float s = 0.0f;
  for (int i = tid; i < n; i += 256) s += parts[i];   // fixed order: deterministic
  red[tid] = s;
  __syncthreads();
#pragma unroll
  for (int k = 128; k > 0; k >>= 1) {
    if (tid < k) red[tid] += red[tid + k];
    __syncthreads();
  }
  if (tid == 0) {
    const float invN = 1.0f / (float)((size_t)NPLANES * IMG_H * IMG_W);
    out[0] = 1.0f - red[0] * invN;
  }
}

extern "C" void kernel_launch(void* const* d_in, const int* in_sizes, int n_in,
                              void* d_out, int out_size, void* d_ws, size_t ws_size,
                              hipStream_t stream) {
  const float* img1 = (const float*)d_in[0];
  const float* img2 = (const float*)d_in[1];
  const float* win  = (const float*)d_in[2];
  float* out   = (float*)d_out;
  float* parts = (float*)d_ws;

  dim3 grid(IMG_W / TILE_W, IMG_H / TILE_H, NPLANES);  // 30 x 34 x 12
  ssim_tile_kernel<<<grid, 256, 0, stream>>>(img1, img2, win, parts);

  int nparts = (IMG_W / TILE_W) * (IMG_H / TILE_H) * NPLANES;
  ssim_reduce_kernel<<<1, 256, 0, stream>>>(parts, nparts, out);
}